// EMAMemory_39891656245789
// MI455X (gfx1250) — compile-verified
//
#include <hip/hip_runtime.h>

// EMA scan over T via chunked-scan-as-GEMM on V_WMMA_F32_16X16X4_F32.
// out[b,0]=x[b,0]; out[b,t]=0.99*out[b,t-1]+0.01*x[b,t]  (per chain n*D+d)
//
// Per wave: 16 contiguous chains, 8 sequential chunks of 16 timesteps.
// Chunk:  D(16x16) = W(16x16, lower-tri consts) * X(16x16) + C(carry)
// realized as 4 chained K=4 fp32 WMMAs. Memory-bound: 1 load + 1 store
// per element. All W coefficients are baked at compile time into a
// __constant__ table (no powf / transcendentals in the kernel).

typedef float v2f __attribute__((ext_vector_type(2)));
typedef float v8f __attribute__((ext_vector_type(8)));

#define EMA_B   4
#define EMA_T   128
#define EMA_CH  (256 * 768)   // chains per batch = N*D = 196608
#define TILE_N  16
#define CHUNK_T 16

// ---------------- compile-time coefficient tables -----------------------
struct EmaTab {
  float wstd[16][16];          // W[i][j] = (j<=i) ? 0.01*0.99^(i-j) : 0
  float wfirst[16][16];        // same, but col 0 = 0.99^i  (out[0]=x[0])
  alignas(32) float pwtab[2][8]; // pwtab[hi][v] = 0.99^(v+8*hi+1)
};

constexpr float pow99(int e) {
  float r = 1.0f;
  for (int t = 0; t < e; ++t) r *= 0.99f;
  return r;
}

constexpr EmaTab make_tab() {
  EmaTab t{};
  for (int i = 0; i < 16; ++i)
    for (int j = 0; j < 16; ++j) {
      const float w = (j <= i) ? 0.01f * pow99(i - j) : 0.0f;
      t.wstd[i][j]   = w;
      t.wfirst[i][j] = (j == 0) ? pow99(i) : w;
    }
  for (int h = 0; h < 2; ++h)
    for (int v = 0; v < 8; ++v)
      t.pwtab[h][v] = pow99(v + 8 * h + 1);
  return t;
}

__constant__ EmaTab TAB = make_tab();

// ---------------- WMMA helper -------------------------------------------
__device__ __forceinline__ v8f wmma4(v2f a, v2f b, v8f c) {
  // 8 args: (neg_a, A, neg_b, B, c_mod, C, reuse_a, reuse_b)
  return __builtin_amdgcn_wmma_f32_16x16x4_f32(false, a, false, b,
                                               (short)0, c, false, false);
}

__global__ __launch_bounds__(256)
void ema_wmma_scan_kernel(const float* __restrict__ in, float* __restrict__ out) {
  const int lane = threadIdx.x & 31;
  const int hi   = lane >> 4;     // half-wave select
  const int col  = lane & 15;     // chain column; also A-matrix row M
  const long long CH = EMA_CH;

  const int wave = (int)((blockIdx.x * blockDim.x + threadIdx.x) >> 5);
  const int tilesPerBatch = EMA_CH / TILE_N;              // 12288
  const int b  = wave / tilesPerBatch;
  const long long c0 = (long long)(wave % tilesPerBatch) * TILE_N;

  // ---- coefficient fetch: 9 vector loads from __constant__ table ------
  // f32 A layout: lanes 0-15 hold K = 4k+0 / 4k+1 (VGPR0/VGPR1),
  //               lanes 16-31 hold K = 4k+2 / 4k+3; M = lane&15.
  // jx = 4k+2*hi is even -> (W[i][jx], W[i][jx+1]) is an aligned b64 pair.
  v2f Astd[4], Afirst[4];
#pragma unroll
  for (int k = 0; k < 4; ++k) {
    const int jx = 4 * k + 2 * hi;
    Astd[k]   = *(const v2f*)&TAB.wstd[col][jx];
    Afirst[k] = *(const v2f*)&TAB.wfirst[col][jx];
  }
  // carry weights: C[v] = pwv[v] * m, with M = v + 8*hi  (C/D VGPR layout)
  const v8f pwv = *(const v8f*)&TAB.pwtab[hi][0];

  const float* __restrict__ ibase = in  + (long long)b * EMA_T * CH + c0 + col;
  float*       __restrict__ obase = out + (long long)b * EMA_T * CH + c0 + col;

  v8f acc;

  // ---- chunk 0: no carry, W_first ------------------------------------
  {
    v2f Bv[4];
#pragma unroll
    for (int k = 0; k < 4; ++k) {
      const int r0 = 4 * k + 2 * hi;          // B K-rows for this half-wave
      Bv[k].x = ibase[(long long)(r0    ) * CH];
      Bv[k].y = ibase[(long long)(r0 + 1) * CH];
    }
    v8f c = {0.f, 0.f, 0.f, 0.f, 0.f, 0.f, 0.f, 0.f};
    acc = wmma4(Afirst[0], Bv[0], c);
    acc = wmma4(Afirst[1], Bv[1], acc);
    acc = wmma4(Afirst[2], Bv[2], acc);
    acc = wmma4(Afirst[3], Bv[3], acc);
#pragma unroll
    for (int v = 0; v < 8; ++v)
      obase[(long long)(v + 8 * hi) * CH] = acc[v];
  }

  // ---- chunks 1..7: carry = previous chunk's last row -----------------
  for (int q = 1; q < EMA_T / CHUNK_T; ++q) {
    const long long t0 = (long long)q * CHUNK_T;
    // D[M=15][N=col] lives in VGPR7 of lanes 16-31 at N = lane-16
    const float m = __shfl(acc[7], 16 + col, 32);

    v2f Bv[4];
#pragma unroll
    for (int k = 0; k < 4; ++k) {
      const int r0 = 4 * k + 2 * hi;
      Bv[k].x = ibase[(t0 + r0    ) * CH];
      Bv[k].y = ibase[(t0 + r0 + 1) * CH];
    }
    v8f c = pwv * m;

    acc = wmma4(Astd[0], Bv[0], c);
    acc = wmma4(Astd[1], Bv[1], acc);
    acc = wmma4(Astd[2], Bv[2], acc);
    acc = wmma4(Astd[3], Bv[3], acc);
#pragma unroll
    for (int v = 0; v < 8; ++v)
      obase[(t0 + v + 8 * hi) * CH] = acc[v];
  }
}

extern "C" void kernel_launch(void* const* d_in, const int* in_sizes, int n_in,
                              void* d_out, int out_size, void* d_ws, size_t ws_size,
                              hipStream_t stream) {
  (void)in_sizes; (void)n_in; (void)out_size; (void)d_ws; (void)ws_size;
  const float* in  = (const float*)d_in[0];   // visual_features, fp32
  float*       out = (float*)d_out;           // per-frame EMA states, fp32

  const int waves   = EMA_B * (EMA_CH / TILE_N);  // 49152 waves, EXEC all-ones
  const int threads = 256;                        // 8 wave32 per block
  const int blocks  = waves * 32 / threads;       // 6144
  hipLaunchKernelGGL(ema_wmma_scan_kernel, dim3(blocks), dim3(threads), 0, stream,
                     in, out);
}